// BaseGCL_20684562498312
// MI455X (gfx1250) — compile-verified
//
#include <hip/hip_runtime.h>
#include <cstdint>

#define N_USERS 100000
#define N_ITEMS 200000
#define N_NODES (N_USERS + N_ITEMS)
#define EMB 64
#define NNZ 4000000
#define GROUPS (NNZ / 2)          /* 2 nnz per wave-group -> 2,000,000 groups */
#define SPMM_BLOCKS 4096
#define NWAVES (SPMM_BLOCKS * 8)
#define TOTAL4 ((size_t)N_NODES * EMB / 4)

// Every wave must own at least 4 groups so the 3-deep async priming is valid.
static_assert((size_t)4 * NWAVES <= GROUPS, "async pipeline priming unsafe");

// COO SpMM: Y[row] += val * X[col]. One wave handles 2 nnz per group (16
// lanes x float4 each). Row gathers are staged LDS-side with gfx1250 async
// loads (global_load_async_to_lds_b128, ASYNCcnt) in a 4-slot ring: 3 gathers
// in flight per wave at steady state (s_wait_asynccnt 0x3), zero VGPR cost.
// Scatter uses native non-returning f32 global atomics (global_atomic_add_f32).
__global__ __launch_bounds__(256) void spmm_kernel(
    const int* __restrict__ rows, const int* __restrict__ cols,
    const float* __restrict__ vals, const float* __restrict__ X,
    float* __restrict__ Y) {
  __shared__ float lds[4][8][2][EMB];  // [slot][wave][sub-nnz][64] = 16 KB

  const int tid  = threadIdx.x;
  const int wave = tid >> 5;
  const int lane = tid & 31;
  const int sub  = lane >> 4;   // which of the 2 nnz in this wave's group
  const int li   = lane & 15;   // 16 lanes cover 64 floats as float4 each
  const int nwaves = gridDim.x * 8;
  const int wid    = blockIdx.x * 8 + wave;

  // Low 32 bits of a generic shared pointer == LDS byte offset (aperture rule).
  uint32_t ldsb[4];
  ldsb[0] = (uint32_t)(uintptr_t)&lds[0][wave][sub][li * 4];
  ldsb[1] = (uint32_t)(uintptr_t)&lds[1][wave][sub][li * 4];
  ldsb[2] = (uint32_t)(uintptr_t)&lds[2][wave][sub][li * 4];
  ldsb[3] = (uint32_t)(uintptr_t)&lds[3][wave][sub][li * 4];

  auto issue = [&](int g, uint32_t ldsoff) {
    const int nz = g * 2 + sub;
    const int c  = cols[nz];
    const uint64_t gaddr = (uint64_t)(uintptr_t)(X + (size_t)c * EMB + li * 4);
    asm volatile("global_load_async_to_lds_b128 %0, %1, off"
                 :: "v"(ldsoff), "v"(gaddr)
                 : "memory");
  };

  // Prime the ring with 3 gathers (every wave owns >= 4 groups; see assert).
  issue(wid + 0 * nwaves, ldsb[0]);
  issue(wid + 1 * nwaves, ldsb[1]);
  issue(wid + 2 * nwaves, ldsb[2]);

  int k = 0;
  for (int g = wid; g < GROUPS; g += nwaves, ++k) {
    const int gn = g + 3 * nwaves;
    if (gn < GROUPS) {                 // uniform branch: EXEC stays full
      issue(gn, ldsb[(k + 3) & 3]);
      asm volatile("s_wait_asynccnt 0x3" ::: "memory");  // oldest slot ready
    } else {
      asm volatile("s_wait_asynccnt 0x0" ::: "memory");  // drain tail
    }

    const int   slot = k & 3;
    const int   nz = g * 2 + sub;
    const int   r  = rows[nz];
    const float v  = vals[nz];
    const float4 xv = *(const float4*)&lds[slot][wave][sub][li * 4];

    float* dst = Y + (size_t)r * EMB + li * 4;
    unsafeAtomicAdd(dst + 0, v * xv.x);
    unsafeAtomicAdd(dst + 1, v * xv.y);
    unsafeAtomicAdd(dst + 2, v * xv.z);
    unsafeAtomicAdd(dst + 3, v * xv.w);
  }
}

// X = concat(user,item); OUT = 0.25*emb (layer-0 accumulator); Y = 0
__global__ __launch_bounds__(256) void init_kernel(
    const float* __restrict__ user, const float* __restrict__ item,
    float* __restrict__ X, float* __restrict__ Y, float* __restrict__ OUT) {
  size_t i4 = (size_t)blockIdx.x * blockDim.x + threadIdx.x;
  if (i4 >= TOTAL4) return;
  size_t i = i4 * 4;
  const float* src = (i < (size_t)N_USERS * EMB)
                         ? (user + i)
                         : (item + (i - (size_t)N_USERS * EMB));
  float4 v = *(const float4*)src;
  *(float4*)(X + i) = v;
  float4 s;
  s.x = 0.25f * v.x; s.y = 0.25f * v.y; s.z = 0.25f * v.z; s.w = 0.25f * v.w;
  *(float4*)(OUT + i) = s;
  float4 z = make_float4(0.f, 0.f, 0.f, 0.f);
  *(float4*)(Y + i) = z;
}

// OUT += 0.25*Ynew; optionally zero Xold (it becomes next layer's target).
__global__ __launch_bounds__(256) void accum_zero_kernel(
    const float* __restrict__ Ynew, float* __restrict__ OUT,
    float* __restrict__ Xold, int do_zero) {
  size_t i4 = (size_t)blockIdx.x * blockDim.x + threadIdx.x;
  if (i4 >= TOTAL4) return;
  size_t i = i4 * 4;
  float4 y = *(const float4*)(Ynew + i);
  float4 o = *(float4*)(OUT + i);
  o.x += 0.25f * y.x; o.y += 0.25f * y.y;
  o.z += 0.25f * y.z; o.w += 0.25f * y.w;
  *(float4*)(OUT + i) = o;
  if (do_zero) {
    float4 z = make_float4(0.f, 0.f, 0.f, 0.f);
    *(float4*)(Xold + i) = z;
  }
}

extern "C" void kernel_launch(void* const* d_in, const int* in_sizes, int n_in,
                              void* d_out, int out_size, void* d_ws, size_t ws_size,
                              hipStream_t stream) {
  (void)in_sizes; (void)n_in; (void)out_size; (void)ws_size;
  const float* user = (const float*)d_in[0];
  const float* item = (const float*)d_in[1];
  const int*   rows = (const int*)d_in[2];
  const int*   cols = (const int*)d_in[3];
  const float* vals = (const float*)d_in[4];

  float* OUT = (float*)d_out;
  float* X = (float*)d_ws;                       // 76.8 MB
  float* Y = X + (size_t)N_NODES * EMB;          // 76.8 MB (both L2-resident)

  const int eb = (int)((TOTAL4 + 255) / 256);    // 18750 blocks
  init_kernel<<<eb, 256, 0, stream>>>(user, item, X, Y, OUT);

  for (int l = 0; l < 3; ++l) {
    spmm_kernel<<<SPMM_BLOCKS, 256, 0, stream>>>(rows, cols, vals, X, Y);
    accum_zero_kernel<<<eb, 256, 0, stream>>>(Y, OUT, X, (l < 2) ? 1 : 0);
    float* t = X; X = Y; Y = t;
  }
}